// LinearChainCRF_40424232190446
// MI455X (gfx1250) — compile-verified
//
#include <hip/hip_runtime.h>
#include <cstdint>

#define CRF_B 256
#define CRF_T 2048
#define CRF_L 48

typedef float v2f __attribute__((ext_vector_type(2)));
typedef float v8f __attribute__((ext_vector_type(8)));

// broadcast/read a float from another lane (wave32, EXEC all-ones in our use)
__device__ __forceinline__ float lane_read(float v, int srcLane) {
  return __int_as_float(
      __builtin_amdgcn_ds_bpermute(srcLane << 2, __float_as_int(v)));
}

// Raw hardware transcendentals (no denormal-safe expansion):
// exp2 underflow->0 is exactly what logsumexp wants; log input is
// always normal (w in [0.9, 53]).
__device__ __forceinline__ float fast_exp2(float x) {
  return __builtin_amdgcn_exp2f(x);
}
__device__ __forceinline__ float fast_log2(float x) {
  return __builtin_amdgcn_logf(x);
}

// XOR-swizzled index into alphaT[48][16] (state-major, chain-minor):
// element (j, m) -> j*16 + (m ^ (j&12)).  Makes alpha-new stores and
// A-fragment loads LDS-bank conflict free.
__device__ __forceinline__ int sidx(int j, int m) {
  return j * 16 + (m ^ (j & 12));
}

// ---------------------------------------------------------------------------
// Forward recursion: 16 chains per wave, one wave per block, barrier-free.
// alphaT kept in log space in LDS (transposed);  each step:
//   load row values once in A-fragment order (the two halves' k-sets are
//   complementary, so their union covers all 48 states);
//   M = rowmax  (24-wide register max + one cross-half bpermute);
//   v = exp2(alpha*log2e - M*log2e)  (raw v_exp_f32);
//   w = v * E   (36x V_WMMA_F32_16X16X4_F32, E = exp(trans) in regs);
//   alpha' = ln2*log2(w) + M + em[t]  (emissions software-pipelined in regs)
// ---------------------------------------------------------------------------
__global__ __launch_bounds__(32) void crf_forward_kernel(
    const float* __restrict__ em, const float* __restrict__ trans,
    float* __restrict__ partition)
{
  __shared__ float alphaT[48 * 16];

  const int lane = threadIdx.x;
  const int nlo  = lane & 15;
  const int half = lane >> 4;          // 0: lanes 0-15, 1: lanes 16-31
  const int b0   = blockIdx.x * 16;

  const float LOG2E = 1.4426950408889634f;
  const float LN2   = 0.6931471805599453f;

  // Preload B fragments of E = exp(transitions): 3 N-tiles x 12 K-slices.
  // B 4x16 layout: N = lane&15, K = vgpr + 2*half.
  v2f bf[3][12];
#pragma unroll
  for (int nt = 0; nt < 3; ++nt) {
#pragma unroll
    for (int ks = 0; ks < 12; ++ks) {
      const int n = nt * 16 + nlo;
      const int k = ks * 4 + 2 * half;
      bf[nt][ks].x = __expf(trans[k * CRF_L + n]);
      bf[nt][ks].y = __expf(trans[(k + 1) * CRF_L + n]);
    }
  }

  // Per-lane emission base pointers for the D-layout rows this lane writes
  // (m = r + 8*half), pre-offset by column nlo.
  const float* embase[8];
#pragma unroll
  for (int r = 0; r < 8; ++r)
    embase[r] = em + (size_t)(b0 + r + 8 * half) * CRF_T * CRF_L + nlo;

  // Chain owned by this lane for init / A-fragments / prefetch.
  const float* ebm = em + (size_t)(b0 + nlo) * CRF_T * CRF_L;

  // alpha0[j][m] = trans[0][j] + em[b0+m, 0, j]; flat idx = it*32+lane gives
  // m = lane&15 (constant per lane), j = 2*it + half.
#pragma unroll
  for (int it = 0; it < 24; ++it) {
    const int j = 2 * it + half;
    alphaT[sidx(j, nlo)] = trans[j] + ebm[j];
  }

  // Pipeline: preload emissions for t = 1.
  float ereg[24];
#pragma unroll
  for (int nt = 0; nt < 3; ++nt)
#pragma unroll
    for (int r = 0; r < 8; ++r)
      ereg[nt * 8 + r] = embase[r][CRF_L + nt * 16];

  for (int t = 1; t < CRF_T; ++t) {
    // ---- load this lane's 24 row values (A-fragment k-pattern) once
    float araw[24];
#pragma unroll
    for (int ks = 0; ks < 12; ++ks) {
      const int k = ks * 4 + 2 * half;
      araw[2 * ks]     = alphaT[sidx(k, nlo)];
      araw[2 * ks + 1] = alphaT[sidx(k + 1, nlo)];
    }

    // ---- row max: 24-wide register max; the complementary half holds the
    //      other 24 states of the same chain, so one bpermute completes it.
    float mx = araw[0];
#pragma unroll
    for (int i = 1; i < 24; ++i) mx = fmaxf(mx, araw[i]);
    mx = fmaxf(mx, lane_read(mx, lane ^ 16));  // M[nlo] in every lane

    // redistribute to D-row order: mr[r] = M[r + 8*half]
    float mr[8];
#pragma unroll
    for (int r = 0; r < 8; ++r) mr[r] = lane_read(mx, r + 8 * half);

    // ---- emissions for t+1 (consumed at the end of the next iteration)
    const int tn   = (t + 1 < CRF_T) ? (t + 1) : t;
    const int toff = tn * CRF_L;
    float enext[24];
#pragma unroll
    for (int nt = 0; nt < 3; ++nt)
#pragma unroll
      for (int r = 0; r < 8; ++r)
        enext[nt * 8 + r] = embase[r][toff + nt * 16];

    // prefetch t+2 emission lines for this lane's chain
    if (t + 2 < CRF_T) {
      const int pf = (t + 2) * CRF_L;
      __builtin_prefetch(&ebm[pf], 0, 0);
      __builtin_prefetch(&ebm[pf + 32], 0, 0);
    }

    // ---- A fragments: v = exp2(alpha*log2e - M*log2e), raw v_exp_f32
    const float nM = -mx * LOG2E;
    v2f af[12];
#pragma unroll
    for (int ks = 0; ks < 12; ++ks) {
      af[ks].x = fast_exp2(fmaf(araw[2 * ks],     LOG2E, nM));
      af[ks].y = fast_exp2(fmaf(araw[2 * ks + 1], LOG2E, nM));
    }

    // ---- w = v * E : 3 N-tiles, accumulate over 12 K-slices each
    v8f acc[3];
#pragma unroll
    for (int nt = 0; nt < 3; ++nt) {
      v8f c = {};
#pragma unroll
      for (int ks = 0; ks < 12; ++ks) {
        c = __builtin_amdgcn_wmma_f32_16x16x4_f32(
            /*neg_a=*/false, af[ks], /*neg_b=*/false, bf[nt][ks],
            /*c_mod=*/(short)0, c, /*reuse_a=*/false, /*reuse_b=*/false);
      }
      acc[nt] = c;
    }

    // ---- alpha' = ln2*log2(w) + M[m] + em[t];  D layout M = r+8*half,
    //      N = lane&15.  Raw v_log_f32.
#pragma unroll
    for (int nt = 0; nt < 3; ++nt) {
#pragma unroll
      for (int r = 0; r < 8; ++r) {
        const int m = r + 8 * half;
        const int j = nt * 16 + nlo;
        const float base = mr[r] + ereg[nt * 8 + r];
        alphaT[sidx(j, m)] = fmaf(fast_log2(acc[nt][r]), LN2, base);
      }
    }

#pragma unroll
    for (int i = 0; i < 24; ++i) ereg[i] = enext[i];
  }

  // partition[b] = logsumexp_j(alpha[j] + trans[j][L-1])  (one-time epilogue)
  if (lane < 16) {
    float mx = -3.4e38f;
#pragma unroll
    for (int j = 0; j < CRF_L; ++j)
      mx = fmaxf(mx, alphaT[sidx(j, lane)] + trans[j * CRF_L + CRF_L - 1]);
    float s = 0.f;
#pragma unroll
    for (int j = 0; j < CRF_L; ++j)
      s += fast_exp2(LOG2E * (alphaT[sidx(j, lane)] + trans[j * CRF_L + CRF_L - 1] - mx));
    partition[b0 + lane] = fmaf(fast_log2(s), LN2, mx);
  }
}

// ---------------------------------------------------------------------------
// Gold scores: one block per batch element, strided gather + LDS reduction.
// ---------------------------------------------------------------------------
__global__ __launch_bounds__(256) void crf_scores_kernel(
    const float* __restrict__ em, const int* __restrict__ tags,
    const float* __restrict__ trans, float* __restrict__ scores)
{
  const int b   = blockIdx.x;
  const int tid = threadIdx.x;
  const int* tg   = tags + (size_t)b * CRF_T;
  const float* eb = em + (size_t)b * CRF_T * CRF_L;

  float s = 0.f;
  for (int t = tid; t < CRF_T; t += 256) {
    const int ct = tg[t];
    s += eb[(size_t)t * CRF_L + ct];
    if (t > 0) s += trans[tg[t - 1] * CRF_L + ct];
  }

  __shared__ float red[256];
  red[tid] = s;
  __syncthreads();
  for (int off = 128; off > 0; off >>= 1) {
    if (tid < off) red[tid] += red[tid + off];
    __syncthreads();
  }
  if (tid == 0)
    scores[b] = red[0] + trans[tg[0]] + trans[tg[CRF_T - 1] * CRF_L + CRF_L - 1];
}

// ---------------------------------------------------------------------------
// Final reduction: out = sum_b (partition[b] - scores[b])
// ---------------------------------------------------------------------------
__global__ __launch_bounds__(256) void crf_reduce_kernel(
    const float* __restrict__ partition, const float* __restrict__ scores,
    float* __restrict__ out)
{
  __shared__ float red[256];
  const int tid = threadIdx.x;
  red[tid] = partition[tid] - scores[tid];
  __syncthreads();
  for (int off = 128; off > 0; off >>= 1) {
    if (tid < off) red[tid] += red[tid + off];
    __syncthreads();
  }
  if (tid == 0) out[0] = red[0];
}

extern "C" void kernel_launch(void* const* d_in, const int* in_sizes, int n_in,
                              void* d_out, int out_size, void* d_ws, size_t ws_size,
                              hipStream_t stream) {
  (void)in_sizes; (void)n_in; (void)out_size; (void)ws_size;

  const float* em    = (const float*)d_in[0];   // (256, 2048, 48) f32
  const int*   tags  = (const int*)d_in[1];     // (256, 2048) i32
  const float* trans = (const float*)d_in[2];   // (48, 48) f32
  float* out = (float*)d_out;                   // scalar f32

  float* partition = (float*)d_ws;              // 256 f32
  float* scores    = partition + CRF_B;         // 256 f32

  crf_scores_kernel<<<CRF_B, 256, 0, stream>>>(em, tags, trans, scores);
  crf_forward_kernel<<<CRF_B / 16, 32, 0, stream>>>(em, trans, partition);
  crf_reduce_kernel<<<1, CRF_B, 0, stream>>>(partition, scores, out);
}